// DualBiPlane_1778116460857
// MI455X (gfx1250) — compile-verified
//
#include <hip/hip_runtime.h>

// DualBiPlane gather kernel for MI455X (gfx1250, wave32).
//
// Mapping: 8 lanes per point. sub = tid&7:
//   sub 0..3 -> Fxy plane, channel block sub*4 .. sub*4+3
//   sub 4..7 -> Fuv plane, channel block (sub-4)*4 .. +3
// Output layout is [point][32] with [0:16]=lat_xy, [16:32]=lat_uv, so the
// float4 output slot for a lane is exactly `tid` -> fully coalesced b128
// stores, 512 contiguous bytes per wave32.
//
// All plane selection is done with integer bit-masking on loaded values so
// the compiler cannot sink loads into divergent EXEC branches: every wave
// has lanes on both planes, so branching would execute both sides anyway.

typedef float f32x4 __attribute__((ext_vector_type(4)));
typedef float f32x2 __attribute__((ext_vector_type(2)));

#define RES 400
// floats per (m,i) row = RES*16 ; in float4 units: RES*4
#define ROW4   (RES * 4)
#define PLANE4 (RES * RES * 4)

__global__ __launch_bounds__(256) void dualbiplane_gather_kernel(
    const int*   __restrict__ m,
    const float* __restrict__ h,
    const float* __restrict__ u,
    const float* __restrict__ v,
    const float* __restrict__ Fxy,
    const float* __restrict__ Fuv,
    float*       __restrict__ out,
    int n_pts)
{
    const int tid = blockIdx.x * blockDim.x + threadIdx.x;
    const int pt  = tid >> 3;
    if (pt >= n_pts) return;

    const int sub   = tid & 7;
    const int plane = sub >> 2;     // 0 = Fxy, 1 = Fuv
    const int c4    = sub & 3;      // float4 channel block within plane

    // --- per-point scalars, loaded unconditionally (lanes share lines) ---
    const int   mi  = m[pt];
    const f32x2 hxy = ((const f32x2*)h)[pt];     // b64 load of (hx, hy)
    const float uu  = u[pt];
    const float vv  = v[pt];

    // Candidate coordinates for both planes.
    // (h+1)/2*400 == (h+1)*200 bit-exactly (mult by 0.5 is exact).
    const float x0 = (hxy.x + 1.0f) * 200.0f;
    const float y0 = (hxy.y + 1.0f) * 200.0f;
    const float x1 = uu * (float)RES;
    const float y1 = vv * (float)RES;

    // Branchless select via bit masks: consumes both candidates, so the
    // compiler cannot sink the u/v/h loads into divergent branches.
    const unsigned pm = 0u - (unsigned)plane;    // 0x00000000 or 0xFFFFFFFF
    float x = __uint_as_float((__float_as_uint(x1) &  pm) |
                              (__float_as_uint(x0) & ~pm));
    float y = __uint_as_float((__float_as_uint(y1) &  pm) |
                              (__float_as_uint(y0) & ~pm));

    // Per-lane table base pointer, same bit-select trick (64-bit).
    const unsigned long long pm64 = 0ull - (unsigned long long)plane;
    const float* __restrict__ F = (const float*)(
        ((unsigned long long)(uintptr_t)Fuv &  pm64) |
        ((unsigned long long)(uintptr_t)Fxy & ~pm64));

    // Reference edge clamp: where(ind == RES, RES-1, ind)  -> v_cndmask
    if (x == (float)RES) x = (float)(RES - 1);
    if (y == (float)RES) y = (float)(RES - 1);

    const float i1f = floorf(x);
    const float j1f = floorf(y);
    const int   i1  = (int)i1f;
    const int   j1  = (int)j1f;
    const int   i2  = (i1 + 1 == RES) ? 0 : i1 + 1;   // (i1+1) % RES
    const int   j2  = (j1 + 1 == RES) ? 0 : j1 + 1;   // (j1+1) % RES
    const float ir  = x - i1f;
    const float jr  = y - j1f;

    // --- 4 corner gathers of 16B each; tables (82MB total) stay resident
    // in the 192MB L2 with default RT temporal policy. ---
    const f32x4* __restrict__ F4 = (const f32x4*)F;
    const int mbase = mi * PLANE4;
    const int r1 = mbase + i1 * ROW4;
    const int r2 = mbase + i2 * ROW4;

    const f32x4 g00 = F4[r1 + j1 * 4 + c4];
    const f32x4 g10 = F4[r2 + j1 * 4 + c4];
    const f32x4 g01 = F4[r1 + j2 * 4 + c4];
    const f32x4 g11 = F4[r2 + j2 * 4 + c4];

    // Same algebraic form as the reference for numeric agreement.
    const float wir = 1.0f - ir;
    const float wjr = 1.0f - jr;
    const f32x4 a   = g00 * wir + g10 * ir;
    const f32x4 b   = g01 * wir + g11 * ir;
    const f32x4 res = a * wjr + b * jr;

    // Non-temporal b128 store (gfx1250 th:TH_NT): the 256MB output stream
    // must not evict the L2-resident feature tables.
    __builtin_nontemporal_store(res, (f32x4*)out + tid);
}

extern "C" void kernel_launch(void* const* d_in, const int* in_sizes, int n_in,
                              void* d_out, int out_size, void* d_ws, size_t ws_size,
                              hipStream_t stream) {
    const int*   m   = (const int*)  d_in[0];
    const float* h   = (const float*)d_in[1];
    const float* u   = (const float*)d_in[2];
    const float* v   = (const float*)d_in[3];
    const float* Fxy = (const float*)d_in[4];
    const float* Fuv = (const float*)d_in[5];
    float* out = (float*)d_out;

    const int n_pts = in_sizes[0];                 // N points
    const long long total = (long long)n_pts * 8;  // 8 lanes per point
    const int block = 256;                         // 8 wave32s per block
    const unsigned grid = (unsigned)((total + block - 1) / block);

    dualbiplane_gather_kernel<<<grid, block, 0, stream>>>(m, h, u, v, Fxy, Fuv,
                                                          out, n_pts);
}